// FFMLayer_78323023610367
// MI455X (gfx1250) — compile-verified
//
#include <hip/hip_runtime.h>

typedef __attribute__((ext_vector_type(2))) float v2f;
typedef __attribute__((ext_vector_type(8))) float v8f;

#define B_ROWS 16384
#define P_DIM  512
#define F_DIM  16
#define K_DIM  8

#define ROWS_PER_BLOCK 64     // 4 waves * 16 rows -> 130KB LDS -> 2 WGs/WGP
#define THREADS        128    // 4 waves (wave32)

// ---------------------------------------------------------------------------
// Kernel 1: S0[i,j] = 0.5 * sum_k v[i,f2f[j],k] * v[j,f2f[i],k]  (0 on diag)
// Folding the 0.5 and the zero diagonal makes rp[b] = x_b^T S0 x_b exactly.
// ---------------------------------------------------------------------------
__global__ void ffm_build_s0(const float* __restrict__ v,
                             const int* __restrict__ f2f,
                             float* __restrict__ s0) {
    int idx = blockIdx.x * blockDim.x + threadIdx.x;   // 0 .. 512*512-1
    int i = idx >> 9;
    int j = idx & (P_DIM - 1);
    int fi = f2f[i];
    int fj = f2f[j];
    const float* vi = v + (i * F_DIM + fj) * K_DIM;
    const float* vj = v + (j * F_DIM + fi) * K_DIM;
    float s = 0.f;
#pragma unroll
    for (int k = 0; k < K_DIM; ++k) s += vi[k] * vj[k];
    s0[idx] = (i == j) ? 0.f : 0.5f * s;
}

// ---------------------------------------------------------------------------
// Kernel 2 (fused): out[b] = x_b.w + x_b^T S0 x_b + bias
// Block: 128 threads = 4 waves; block owns 64 rows (16 per wave).
// X tile (64x512 f32 = 128KB) + w (2KB) staged in LDS -> 2 workgroups/WGP
// co-resident (4 waves per SIMD32) to hide the L2 latency of the S0 stream.
// Per wave: 8 groups of 4 N-tiles; one LDS A load feeds 4 independent
// v_wmma_f32_16x16x4_f32 accumulator chains; Y tiles are consumed in
// registers by the fused row-dot (never stored).
// ---------------------------------------------------------------------------
__global__ void ffm_fused_wmma(const float* __restrict__ x,
                               const float* __restrict__ s0,
                               const float* __restrict__ w,
                               const float* __restrict__ bias,
                               float* __restrict__ out) {
    extern __shared__ float xs[];                 // 64*512 X tile + 512 w
    const int tid    = threadIdx.x;               // 0..127
    const int wave   = tid >> 5;                  // 0..3
    const int lane   = tid & 31;                  // wave32
    const int rowBlk = blockIdx.x * ROWS_PER_BLOCK;
    float* const wl  = xs + ROWS_PER_BLOCK * P_DIM;

    // Cooperative 16B-wide staging of the X block (+ w) into LDS.
    {
        const float4* xg = (const float4*)(x + (size_t)rowBlk * P_DIM);
        float4*       xl = (float4*)xs;
        const int nvec = ROWS_PER_BLOCK * P_DIM / 4;   // 8192 float4
        for (int t = tid; t < nvec; t += THREADS) xl[t] = xg[t];
        if (tid < P_DIM / 4) ((float4*)wl)[tid] = ((const float4*)w)[tid];
    }
    const float bias0 = bias[0];
    __syncthreads();

    const int row0 = wave * 16;                   // wave's rows inside LDS tile
    const int m    = lane & 15;                   // M index (A) / N index (B,C,D)
    const int kh   = lane >> 4;                   // half-wave selector

    float acc[8];
#pragma unroll
    for (int r = 0; r < 8; ++r) acc[r] = 0.f;

    for (int ntg = 0; ntg < P_DIM / 64; ++ntg) {  // 8 groups of 4 N-tiles
        const int col0 = ntg * 64;
        v8f c0 = {}, c1 = {}, c2 = {}, c3 = {};
        for (int kb = 0; kb < P_DIM; kb += 4) {
            const int ka = kb + 2 * kh;           // lanes 0-15: K=kb,kb+1 ; 16-31: kb+2,kb+3
            // A: X rows from LDS (32-bit A 16x4 layout) -> ds_load_2addr_b32 (shared by 4 wmma)
            v2f a;
            a.x = xs[(row0 + m) * P_DIM + ka];
            a.y = xs[(row0 + m) * P_DIM + ka + 1];
            // B: S0 rows, lane-contiguous in N -> coalesced global loads (L2-hot)
            const float* s0r0 = s0 + ka * P_DIM + col0 + m;
            const float* s0r1 = s0 + (ka + 1) * P_DIM + col0 + m;
            v2f b0 = { s0r0[0],  s0r1[0]  };
            v2f b1 = { s0r0[16], s0r1[16] };
            v2f b2 = { s0r0[32], s0r1[32] };
            v2f b3 = { s0r0[48], s0r1[48] };
            c0 = __builtin_amdgcn_wmma_f32_16x16x4_f32(false, a, false, b0, (short)0, c0, false, false);
            c1 = __builtin_amdgcn_wmma_f32_16x16x4_f32(false, a, false, b1, (short)0, c1, false, false);
            c2 = __builtin_amdgcn_wmma_f32_16x16x4_f32(false, a, false, b2, (short)0, c2, false, false);
            c3 = __builtin_amdgcn_wmma_f32_16x16x4_f32(false, a, false, b3, (short)0, c3, false, false);
        }
        // Fused row-dot: rp[M] += sum_n Y[M,n] * X[M, col0+n]
        // C/D layout: VGPR r <-> M = r + 8*kh ; N = lane & 15
#pragma unroll
        for (int r = 0; r < 8; ++r) {
            const int M = kh * 8 + r;
            const float* xr = &xs[(row0 + M) * P_DIM + col0 + m];
            acc[r] += c0[r] * xr[0] + c1[r] * xr[16] + c2[r] * xr[32] + c3[r] * xr[48];
        }
    }

    // Fused linear term: each half-wave computes x[row].w for its 8 rows
    // (16 lanes, K stride 16) so it folds into the same 16-lane reduction.
#pragma unroll
    for (int r = 0; r < 8; ++r) {
        const int M = kh * 8 + r;
        const float* xr = &xs[(row0 + M) * P_DIM];
        float s = 0.f;
#pragma unroll 4
        for (int k = m; k < P_DIM; k += 16) s += xr[k] * wl[k];
        acc[r] += s;
    }

    // Reduce across the 16 lanes of each half-wave.
#pragma unroll
    for (int r = 0; r < 8; ++r) {
        float t = acc[r];
        t += __shfl_xor(t, 8, 32);
        t += __shfl_xor(t, 4, 32);
        t += __shfl_xor(t, 2, 32);
        t += __shfl_xor(t, 1, 32);
        acc[r] = t;
    }
    if (m == 0) {
        // lane 0 -> rows row0+0..7 ; lane 16 -> rows row0+8..15
        const int base = rowBlk + row0 + kh * 8;
#pragma unroll
        for (int r = 0; r < 8; ++r) out[base + r] = acc[r] + bias0;
    }
}

// ---------------------------------------------------------------------------
extern "C" void kernel_launch(void* const* d_in, const int* in_sizes, int n_in,
                              void* d_out, int out_size, void* d_ws, size_t ws_size,
                              hipStream_t stream) {
    const float* x    = (const float*)d_in[0];   // (B,P)
    const float* w    = (const float*)d_in[1];   // (P,1)
    const float* v    = (const float*)d_in[2];   // (P,F,K)
    const float* bias = (const float*)d_in[3];   // (1,)
    const int*   f2f  = (const int*)d_in[4];     // (P,)
    float* out = (float*)d_out;                  // (B,1)

    float* s0 = (float*)d_ws;                    // 512*512 f32 = 1 MB scratch

    // 1) Build scaled, zero-diagonal interaction matrix S0.
    ffm_build_s0<<<(P_DIM * P_DIM) / 256, 256, 0, stream>>>(v, f2f, s0);

    // 2) Fully fused quadratic + linear + bias via f32 WMMA.
    //    64 rows/block, LDS = 64x512 X-tile + w = 130 KB -> 2 WGs/WGP.
    ffm_fused_wmma<<<B_ROWS / ROWS_PER_BLOCK, THREADS,
                     (ROWS_PER_BLOCK * P_DIM + P_DIM) * sizeof(float), stream>>>(
        x, s0, w, bias, out);
}